// FMFMNeuron_42262478192715
// MI455X (gfx1250) — compile-verified
//
#include <hip/hip_runtime.h>

// CDNA5 / gfx1250 SNN Leaky kernel.
//  - cur = spikes @ W^T via V_WMMA_F32_16X16X4_F32 (exact fp32 path)
//  - sequential leaky recurrence per batch lane
//  - 64-deep load pipeline (rides the 6-bit LOADcnt ceiling of 63 outstanding)
//  - all inner-loop addressing folded into 24-bit instruction offsets
//  - non-temporal stores for the streaming 128MB output
typedef __attribute__((ext_vector_type(2))) float v2f;
typedef __attribute__((ext_vector_type(8))) float v8f;

__device__ __forceinline__ float swap16(float x) {
    // ds_swizzle_b32 SWAPX16: xor_mask=0x10, and_mask=0x1f -> offset 0x401F
    return __int_as_float(__builtin_amdgcn_ds_swizzle(__float_as_int(x), 0x401F));
}

__global__ __launch_bounds__(256)
void snn_leaky_wmma_kernel(const float* __restrict__ spike,
                           const float* __restrict__ W,
                           float* __restrict__ out) {
    constexpr int T  = 1024;
    constexpr int Bn = 16384;
    constexpr int PF = 64;        // prefetch depth (register ring, 128 VGPRs)

    const int tid  = blockIdx.x * 256 + threadIdx.x;   // batch column
    const int lane = threadIdx.x & 31;
    const bool lo  = (lane < 16);

    const float2* __restrict__ sp   = (const float2*)spike + tid; // (t) -> sp[t*Bn]
    float*        __restrict__ osp  = out + tid;                  // spk_rec
    float*        __restrict__ omem = out + (size_t)T * Bn + tid; // mem_rec

    const float w0 = W[0];
    const float w1 = W[1];

    // A operand (loop invariant): 16x4 fp32 A layout
    //   lanes 0-15: v[0]=K0, v[1]=K1  -> (w0, w1) for every row m
    //   lanes16-31: v[0]=K2, v[1]=K3  -> 0 (unused K)
    v2f A;
    A.x = lo ? w0 : 0.0f;
    A.y = lo ? w1 : 0.0f;
    const v8f Cz = {};

    float mem = 0.0f;
    float rst = 0.0f;   // reset for step t == spike of step t-1 (mem0=0 -> rst0=0)

    // one timestep: consume spikes s, update mem/rst, store spk+mem (NT)
    auto step = [&](float2 s, float* sd, float* md) {
        // B #1: columns = batches waveBase+0..15 (lanes 0-15's spikes)
        v2f B1;
        B1.x = lo ? s.x : 0.0f;   // K0 row = s0
        B1.y = lo ? s.y : 0.0f;   // K1 row = s1
        // B #2: columns = batches waveBase+16..31 (upper lanes' spikes, SWAPX16'd)
        const float sx = swap16(s.x);
        const float sy = swap16(s.y);
        v2f B2;
        B2.x = lo ? sx : 0.0f;
        B2.y = lo ? sy : 0.0f;

        // D[m][n] = cur[n] for every row m => d[0] at lane L holds cur[L%16]
        v8f d1 = __builtin_amdgcn_wmma_f32_16x16x4_f32(
            false, A, false, B1, (short)0, Cz, false, false);
        v8f d2 = __builtin_amdgcn_wmma_f32_16x16x4_f32(
            false, A, false, B2, (short)0, Cz, false, false);
        const float cur = lo ? d1[0] : d2[0];

        // snntorch Leaky, reset_mechanism='subtract', beta=0.9, threshold=1.0
        mem = 0.9f * mem + cur - rst;
        rst = (mem > 1.0f) ? 1.0f : 0.0f;   // spk_t, and reset for t+1
        __builtin_nontemporal_store(rst, sd);
        __builtin_nontemporal_store(mem, md);
    };

    // prime the ring: loads at compile-time offsets i*Bn*8 bytes
    // (max 63*131072 = 8,257,536 B, fits signed 24-bit IOFFSET)
    float2 ring[PF];
#pragma unroll
    for (int i = 0; i < PF; ++i) ring[i] = sp[i * Bn];

    const float2* __restrict__ pf = sp + PF * Bn;

    // steady state: consume chunk, prefetch next chunk (constant-offset loads/stores)
    for (int tc = 0; tc < T - PF; tc += PF) {
#pragma unroll
        for (int j = 0; j < PF; ++j) {
            float2 s = ring[j];
            ring[j]  = pf[j * Bn];                 // offset j*131072 B, constant
            step(s, osp + j * Bn, omem + j * Bn);  // offsets j*65536 B, constant
        }
        pf   += PF * Bn;
        osp  += PF * Bn;
        omem += PF * Bn;
    }

    // final chunk: no prefetch
#pragma unroll
    for (int j = 0; j < PF; ++j) {
        step(ring[j], osp + j * Bn, omem + j * Bn);
    }
}

extern "C" void kernel_launch(void* const* d_in, const int* in_sizes, int n_in,
                              void* d_out, int out_size, void* d_ws, size_t ws_size,
                              hipStream_t stream) {
    (void)in_sizes; (void)n_in; (void)d_ws; (void)ws_size; (void)out_size;
    const float* spike = (const float*)d_in[0];   // (1024, 16384, 2) fp32
    const float* W     = (const float*)d_in[1];   // (1, 2) fp32
    float* out         = (float*)d_out;           // spk_rec ++ mem_rec

    dim3 block(256);                 // 8 wave32s per block
    dim3 grid(16384 / 256);          // 16384 lanes, no tail -> EXEC all ones
    hipLaunchKernelGGL(snn_leaky_wmma_kernel, grid, block, 0, stream,
                       spike, W, out);
}